// MixSelfAttention_88046829568165
// MI455X (gfx1250) — compile-verified
//
#include <hip/hip_runtime.h>
#include <math.h>

typedef __attribute__((ext_vector_type(2))) float v2f;
typedef __attribute__((ext_vector_type(8))) float v8f;

#define LQ    1024
#define HH    8
#define EEC   64
#define NBH   16
#define NTOP  35
#define NPAD  48
#define SCALE 0.125f   // 1/sqrt(64)

__device__ __forceinline__ v8f wmma_f32(v2f a, v2f b, v8f c) {
  // D = A(16x4,f32) x B(4x16,f32) + C(16x16,f32)  -> v_wmma_f32_16x16x4_f32
  return __builtin_amdgcn_wmma_f32_16x16x4_f32(false, a, false, b, (short)0, c,
                                               false, false);
}

// ---------------------------------------------------------------- init amp2_t
__global__ __launch_bounds__(256) void k_init(float* amp2) {
  amp2[blockIdx.x * 256 + threadIdx.x] = 0.f;
}

// ------------------------------------------- [B,L,H,E] -> [B,H,E,L] transpose
__global__ __launch_bounds__(256) void k_transpose(const float* __restrict__ q,
                                                   const float* __restrict__ k,
                                                   float* __restrict__ qT,
                                                   float* __restrict__ kT) {
  int blk = blockIdx.x;                 // 2 * 16 * 16 blocks
  int tc = blk & 15;
  int bh = (blk >> 4) & 15;
  int which = blk >> 8;
  const float* src = which ? k : q;
  float* dst = which ? kT : qT;
  int b = bh >> 3, h = bh & 7;
  __shared__ float tile[64][65];
  int tid = threadIdx.x;
  int e = tid & 63, tl = tid >> 6;
  for (int r = 0; r < 16; ++r) {
    int t = tc * 64 + r * 4 + tl;
    tile[r * 4 + tl][e] = src[(((size_t)b * LQ + t) * HH + h) * EEC + e];
  }
  __syncthreads();
  int t2 = tid & 63, el = tid >> 6;
  for (int r = 0; r < 16; ++r) {
    int e2 = r * 4 + el;
    dst[((size_t)bh * EEC + e2) * LQ + tc * 64 + t2] = tile[t2][e2];
  }
}

// ---------------- circular cross-correlation amplitude^2 (LDS-resident VALU)
// q channel stored duplicated (2048 entries) so the circular read is linear.
__global__ __launch_bounds__(1024) void k_corr(const float* __restrict__ qT,
                                               const float* __restrict__ kT,
                                               float* __restrict__ amp2) {
  int ec = blockIdx.x & 31;             // 32 chunks of 2 channels
  int bh = blockIdx.x >> 5;
  int tau = threadIdx.x;
  __shared__ float qs[2048];
  __shared__ float ks[1024];
  float acc = 0.f;
  for (int c = 0; c < 2; ++c) {
    int e = ec * 2 + c;
    const float* qp = qT + ((size_t)bh * EEC + e) * LQ;
    const float* kp = kT + ((size_t)bh * EEC + e) * LQ;
    if (c == 0) {
      __builtin_prefetch(qT + ((size_t)bh * EEC + e + 1) * LQ, 0, 0);
      __builtin_prefetch(kT + ((size_t)bh * EEC + e + 1) * LQ, 0, 0);
    }
    float qv = qp[tau];
    qs[tau] = qv;
    qs[tau + 1024] = qv;
    ks[tau] = kp[tau];
    __syncthreads();
    const float* qsh = qs + tau;
    float ac = 0.f;
#pragma unroll 8
    for (int t = 0; t < 1024; ++t)
      ac = fmaf(qsh[t], ks[t], ac);
    acc = fmaf(ac, ac, acc);
    __syncthreads();
  }
  atomicAdd(&amp2[bh * LQ + tau], acc);
}

// ------------------------------------------------- |tf_q|^2 over E (no corr)
__global__ __launch_bounds__(256) void k_amp_tf(const float* __restrict__ tfq,
                                                float* __restrict__ amp2) {
  int idx = blockIdx.x * 256 + threadIdx.x;   // 16384
  int bh = idx >> 10, l = idx & 1023;
  int b = bh >> 3, h = bh & 7;
  const float* p = tfq + (((size_t)b * LQ + l) * HH + h) * EEC;
  float s = 0.f;
#pragma unroll 8
  for (int e = 0; e < EEC; ++e) s = fmaf(p[e], p[e], s);
  amp2[idx] = s;
}

// -------------------------------------------- iterative top-35 per (b,h,list)
__global__ __launch_bounds__(256) void k_topk(const float* __restrict__ a_t,
                                              const float* __restrict__ a_tf,
                                              int* __restrict__ top_t,
                                              int* __restrict__ top_tf) {
  int bh = blockIdx.x & 15;
  int which = blockIdx.x >> 4;
  const float* a = (which ? a_tf : a_t) + bh * LQ;
  int* out = (which ? top_tf : top_t) + bh * NPAD;
  __shared__ float vals[1024];
  __shared__ float rv[256];
  __shared__ int ri[256];
  int tid = threadIdx.x;
  for (int i = 0; i < 4; ++i) vals[tid + 256 * i] = a[tid + 256 * i];
  __syncthreads();
  for (int sel = 0; sel < NTOP; ++sel) {
    float bestv = -1.f;
    int besti = 0;
    for (int i = 0; i < 4; ++i) {
      int idx = i * 256 + tid;
      float v = vals[idx];
      if (v > bestv || (v == bestv && idx < besti)) { bestv = v; besti = idx; }
    }
    rv[tid] = bestv; ri[tid] = besti;
    __syncthreads();
    for (int s = 128; s > 0; s >>= 1) {
      if (tid < s) {
        if (rv[tid + s] > rv[tid] ||
            (rv[tid + s] == rv[tid] && ri[tid + s] < ri[tid])) {
          rv[tid] = rv[tid + s]; ri[tid] = ri[tid + s];
        }
      }
      __syncthreads();
    }
    if (tid == 0) { out[sel] = ri[0]; vals[ri[0]] = -1.f; }
    __syncthreads();
  }
  if (tid < NPAD - NTOP) out[NTOP + tid] = 0;   // deterministic padding
}

// ------------------ scores = Q_sel(48x64) x K^T(64x1024), WMMA f32 16x16x4
// Padded A rows (i>=35) use a clamped gather; their C rows are overwritten
// with -inf below, so no zeroing (and no EXEC divergence) is needed.
__global__ __launch_bounds__(256) void k_scores(const float* __restrict__ q,
                                                const float* __restrict__ k,
                                                const float* __restrict__ tfq,
                                                const int* __restrict__ top_t,
                                                const int* __restrict__ top_tf,
                                                float* __restrict__ sc_t,
                                                float* __restrict__ sc_tf) {
  int wg = blockIdx.x * 8 + (threadIdx.x >> 5);   // 6144 waves
  int lane = threadIdx.x & 31;
  int tile = wg % 192;
  int bh = (wg / 192) & 15;
  int phase = wg / (192 * 16);
  int mt = tile / 64, nt = tile % 64;
  int b = bh >> 3, h = bh & 7;
  const float* A_src = phase ? tfq : q;
  const float* B_src = phase ? tfq : k;
  const int* top = (phase ? top_tf : top_t) + bh * NPAD;
  float* out = (phase ? sc_tf : sc_t) + (size_t)bh * NPAD * LQ;

  int m = lane & 15, hx = lane >> 4;
  int i = mt * 16 + m;
  int ic = (i < NTOP) ? i : (NTOP - 1);           // clamped gather index
  int row = top[ic];
  const float* arow = A_src + (((size_t)b * LQ + row) * HH + h) * EEC;
  int mcol = nt * 16 + m;
  const float* brow = B_src + (((size_t)b * LQ + mcol) * HH + h) * EEC;

  v8f c = {};
#pragma unroll
  for (int kb = 0; kb < 16; ++kb) {
    int kk = kb * 4 + 2 * hx;
    v2f a = *(const v2f*)(arow + kk);    // 8B-aligned -> global_load_b64
    v2f bb = *(const v2f*)(brow + kk);
    c = wmma_f32(a, bb, c);
  }
#pragma unroll
  for (int r = 0; r < 8; ++r) {
    int io = mt * 16 + r + 8 * hx;
    float v = (io < NTOP) ? c[r] * SCALE : -__builtin_inff();
    out[(size_t)io * LQ + mcol] = v;
  }
}

// ------------------------------ softmax over the 70 finite entries per column
__global__ __launch_bounds__(256) void k_softmax(float* __restrict__ sc_t,
                                                 float* __restrict__ sc_tf) {
  int bh = blockIdx.x >> 2;
  int l = (blockIdx.x & 3) * 256 + threadIdx.x;
  float* pt = sc_t + (size_t)bh * NPAD * LQ;
  float* pf = sc_tf + (size_t)bh * NPAD * LQ;
  float mx = -__builtin_inff();
  for (int i = 0; i < NTOP; ++i) {
    mx = fmaxf(mx, pt[i * LQ + l]);
    mx = fmaxf(mx, pf[i * LQ + l]);
  }
  float sum = 0.f;
  for (int i = 0; i < NTOP; ++i)
    sum += __expf(pt[i * LQ + l] - mx) + __expf(pf[i * LQ + l] - mx);
  float inv = 1.f / sum;
  for (int i = 0; i < NPAD; ++i) {
    float a = (i < NTOP) ? __expf(pt[i * LQ + l] - mx) * inv : 0.f;
    float bq = (i < NTOP) ? __expf(pf[i * LQ + l] - mx) * inv : 0.f;
    pt[i * LQ + l] = a;
    pf[i * LQ + l] = bq;
  }
}

// --------------- M[i,d] = sum_j W[j, col_i] * V[b,h,j,d]   (WMMA, K=1024)
// Same clamped-gather trick: padded M rows are overwritten with 0 afterwards.
__global__ __launch_bounds__(256) void k_mprep(const float* __restrict__ tw,
                                               const float* __restrict__ vals,
                                               const int* __restrict__ top_t,
                                               const int* __restrict__ top_tf,
                                               float* __restrict__ Mt,
                                               float* __restrict__ Mtf) {
  int wg = blockIdx.x * 8 + (threadIdx.x >> 5);   // 384 waves
  int lane = threadIdx.x & 31;
  int t2 = wg % 24, bh = wg / 24;
  int list = t2 / 12, t3 = t2 % 12;
  int mt = t3 >> 2, nt = t3 & 3;
  const int* top = (list ? top_tf : top_t) + bh * NPAD;
  int m = lane & 15, hx = lane >> 4;
  int i = mt * 16 + m;
  int ic = (i < NTOP) ? i : (NTOP - 1);
  int col = (list ? LQ : 0) + top[ic];
  int b = bh >> 3, h = bh & 7;
  int d = nt * 16 + m;
  const float* vbase = vals + (((size_t)b * LQ) * HH + h) * EEC + d;

  v8f c = {};
  for (int kb = 0; kb < 256; ++kb) {
    int j = kb * 4 + 2 * hx;
    v2f a, bb;
    a.x = tw[(size_t)j * (2 * LQ) + col];
    a.y = tw[(size_t)(j + 1) * (2 * LQ) + col];
    bb.x = vbase[(size_t)j * (HH * EEC)];
    bb.y = vbase[(size_t)(j + 1) * (HH * EEC)];
    c = wmma_f32(a, bb, c);
  }
  float* M = (list ? Mtf : Mt) + (size_t)bh * NPAD * EEC;
#pragma unroll
  for (int r = 0; r < 8; ++r) {
    int io = mt * 16 + r + 8 * hx;
    M[io * EEC + nt * 16 + m] = (io < NTOP) ? c[r] : 0.f;
  }
}

// ----------------------------------------------- bv[d] = sum_j bias[j]*V[j,d]
__global__ __launch_bounds__(64) void k_bv(const float* __restrict__ vals,
                                           const float* __restrict__ bias,
                                           float* __restrict__ bv) {
  int bh = blockIdx.x, d = threadIdx.x;
  int b = bh >> 3, h = bh & 7;
  float s = 0.f;
  for (int j = 0; j < LQ; ++j)
    s = fmaf(bias[j], vals[(((size_t)b * LQ + j) * HH + h) * EEC + d], s);
  bv[bh * EEC + d] = s;
}

// -------- out[l,d] = P(1024x96) x M(96x64) + bv  (WMMA, C seeded with bias·V)
// K-loop split into the Pt/Ma half and the Pf/Mb half: branch-free.
__global__ __launch_bounds__(256) void k_out(const float* __restrict__ sc_t,
                                             const float* __restrict__ sc_tf,
                                             const float* __restrict__ Mt,
                                             const float* __restrict__ Mtf,
                                             const float* __restrict__ bv,
                                             float* __restrict__ out) {
  int wg = blockIdx.x * 8 + (threadIdx.x >> 5);   // 4096 waves
  int lane = threadIdx.x & 31;
  int t2 = wg % 256, bh = wg / 256;
  int mt = t2 >> 2, nt = t2 & 3;
  int b = bh >> 3, h = bh & 7;
  const float* Pt = sc_t + (size_t)bh * NPAD * LQ;
  const float* Pf = sc_tf + (size_t)bh * NPAD * LQ;
  const float* Ma = Mt + (size_t)bh * NPAD * EEC;
  const float* Mb = Mtf + (size_t)bh * NPAD * EEC;
  int m = lane & 15, hx = lane >> 4;
  int l = mt * 16 + m;
  int d = nt * 16 + m;
  float bvv = bv[bh * EEC + d];
  v8f c;
#pragma unroll
  for (int r = 0; r < 8; ++r) c[r] = bvv;

#pragma unroll
  for (int kb = 0; kb < 12; ++kb) {
    int i = kb * 4 + 2 * hx;
    v2f a, bb;
    a.x = Pt[(size_t)i * LQ + l];
    a.y = Pt[(size_t)(i + 1) * LQ + l];
    bb.x = Ma[i * EEC + d];
    bb.y = Ma[(i + 1) * EEC + d];
    c = wmma_f32(a, bb, c);
  }
#pragma unroll
  for (int kb = 0; kb < 12; ++kb) {
    int i = kb * 4 + 2 * hx;
    v2f a, bb;
    a.x = Pf[(size_t)i * LQ + l];
    a.y = Pf[(size_t)(i + 1) * LQ + l];
    bb.x = Mb[i * EEC + d];
    bb.y = Mb[(i + 1) * EEC + d];
    c = wmma_f32(a, bb, c);
  }
#pragma unroll
  for (int r = 0; r < 8; ++r) {
    int lo = mt * 16 + r + 8 * hx;
    out[(((size_t)b * LQ + lo) * HH + h) * EEC + d] = c[r];
  }
}

extern "C" void kernel_launch(void* const* d_in, const int* in_sizes, int n_in,
                              void* d_out, int out_size, void* d_ws, size_t ws_size,
                              hipStream_t stream) {
  (void)in_sizes; (void)n_in; (void)out_size; (void)ws_size;
  const float* tfq = (const float*)d_in[0];
  const float* q   = (const float*)d_in[1];
  const float* k   = (const float*)d_in[2];
  const float* v   = (const float*)d_in[3];
  // d_in[4] = mask (unused by the module)
  const float* tw  = (const float*)d_in[5];
  const float* tb  = (const float*)d_in[6];
  float* out = (float*)d_out;

  float* ws      = (float*)d_ws;
  float* qT      = ws;                       // 1,048,576 f32
  float* kT      = qT + 1048576;             // 1,048,576
  float* sc_t    = kT + 1048576;             //   786,432
  float* sc_tf   = sc_t + 786432;            //   786,432
  float* Mt      = sc_tf + 786432;           //     3,072
  float* Mtf     = Mt + 3072;                //     3,072
  float* bv      = Mtf + 3072;               //     1,024
  float* amp2_t  = bv + 1024;                //    16,384
  float* amp2_tf = amp2_t + 16384;           //    16,384
  int*   top_t   = (int*)(amp2_tf + 16384);  //       768
  int*   top_tf  = top_t + 768;              //       768  (~14.9 MB total)

  k_init     <<<64,  256, 0, stream>>>(amp2_t);
  k_transpose<<<512, 256, 0, stream>>>(q, k, qT, kT);
  k_corr     <<<512, 1024, 0, stream>>>(qT, kT, amp2_t);
  k_amp_tf   <<<64,  256, 0, stream>>>(tfq, amp2_tf);
  k_topk     <<<32,  256, 0, stream>>>(amp2_t, amp2_tf, top_t, top_tf);
  k_scores   <<<768, 256, 0, stream>>>(q, k, tfq, top_t, top_tf, sc_t, sc_tf);
  k_softmax  <<<64,  256, 0, stream>>>(sc_t, sc_tf);
  k_mprep    <<<48,  256, 0, stream>>>(tw, v, top_t, top_tf, Mt, Mtf);
  k_bv       <<<16,  64,  0, stream>>>(v, tb, bv);
  k_out      <<<512, 256, 0, stream>>>(sc_t, sc_tf, Mt, Mtf, bv, out);
}